// ogb_GCN_80762565034380
// MI455X (gfx1250) — compile-verified
//
#include <hip/hip_runtime.h>
#include <hip/hip_bf16.h>
#include <cstdint>
#include <cstddef>

// ---------------------------------------------------------------------------
// GCN (3-layer GraphConv 'both' norm + BN + ReLU) for MI455X / gfx1250.
// GEMMs: V_WMMA_F32_16X16X4_F32 fed from LDS-staged tiles. B is stored
// pair-interleaved so each fragment is a single aligned ds_load_b64 into an
// even VGPR pair (no shuffle movs). Bank-conflict-free strides for 64 banks.
// ---------------------------------------------------------------------------

typedef __attribute__((ext_vector_type(2))) float v2f;
typedef __attribute__((ext_vector_type(8))) float v8f;

#define GCN_IN_FEATS 128
#define GCN_HIDDEN   256
#define GCN_CLASSES  40
#define GCN_BN_EPS   1e-5f

// GEMM macro-tile configuration
#define TM 128            // rows per block (8 waves x 16)
#define TNC 64            // cols per block (4 WMMA tiles of 16)
#define KC 64             // K-chunk staged in LDS
#define A_STRIDE 68       // LDS row stride for A chunk (floats): conflict-free b64 reads
#define B_PSTRIDE 160     // LDS stride per k-pair row of B (floats): conflict-free b64 reads

// ---------------- utility: zero a float buffer ----------------
__global__ void gcn_zero_f32(float* __restrict__ p, int n) {
  int i = blockIdx.x * blockDim.x + threadIdx.x;
  if (i < n) p[i] = 0.0f;
}

// ---------------- degree count + norm ----------------
__global__ void gcn_degree_count(const int* __restrict__ src, const int* __restrict__ dst,
                                 float* __restrict__ degO, float* __restrict__ degI, int E) {
  int e = blockIdx.x * blockDim.x + threadIdx.x;
  if (e < E) {
    atomicAdd(&degO[src[e]], 1.0f);
    atomicAdd(&degI[dst[e]], 1.0f);
  }
}

__global__ void gcn_finalize_norm(float* __restrict__ deg, int n) {
  int i = blockIdx.x * blockDim.x + threadIdx.x;
  if (i < n) deg[i] = rsqrtf(fmaxf(deg[i], 1.0f));
}

// ---------------- WMMA f32 GEMM: C[N,M] = A[N,K] @ B[K,M] ----------------
// Block: 256 threads = 8 waves; computes TM x TNC macro-tile.
// A fragment (ISA 7.12.2, 32-bit A 16x4): lane l, r=l&15, h=l>>4:
//   a = (A[r][k+2h], A[r][k+2h+1])            -> ds_load_b64 from As
// B fragment (4x16): b = (B[k+2h][n], B[k+2h+1][n])
//   BsP pair-interleaved: BsP[q][c] = (B[2q][c], B[2q+1][c]); the pair index
//   a lane needs is q = k/2 + h                -> single ds_load_b64
// C tile: lane holds col n=r, rows 8h+j in acc[j].
// K % KC == 0 (K = 128 or 256). Row/col edges: clamped loads + guarded stores
// keep EXEC all-1s across WMMA (ISA 7.12 requirement).
__global__ __launch_bounds__(256) void gcn_gemm_wmma_f32(
    const float* __restrict__ A, const float* __restrict__ B,
    float* __restrict__ C, int N, int K, int M) {
  __shared__ float As[TM * A_STRIDE];              // 34 KB
  __shared__ float BsP[(KC / 2) * B_PSTRIDE];      // 20 KB

  const int row0 = blockIdx.x * TM;
  const int col0 = blockIdx.y * TNC;
  const int tid  = threadIdx.x;
  const int w    = tid >> 5;                       // wave 0..7
  const int lane = tid & 31;
  const int r    = lane & 15;
  const int h    = lane >> 4;

  v8f acc[4];
  acc[0] = (v8f){}; acc[1] = (v8f){}; acc[2] = (v8f){}; acc[3] = (v8f){};

  const float* Afrag = As + (16 * w + r) * A_STRIDE + 2 * h;
  const float* Bfrag = BsP + h * B_PSTRIDE + 2 * r;

  for (int kb = 0; kb < K; kb += KC) {
    // ---- stage A chunk: TM x KC, float4-vectorized, coalesced ----
    // 128*64/4 = 2048 float4 across 256 threads -> 8 each
#pragma unroll
    for (int i = 0; i < (TM * KC / 4) / 256; ++i) {
      int j = tid + i * 256;                       // float4 index
      int rr = j >> 4;                             // chunk row (16 float4 per row)
      int c4 = j & 15;
      int grow = row0 + rr;
      grow = (grow < N) ? grow : (N - 1);          // clamp (stores are guarded)
      const float4 v = *reinterpret_cast<const float4*>(
          A + (size_t)grow * (size_t)K + (size_t)(kb + c4 * 4));
      *reinterpret_cast<float4*>(&As[rr * A_STRIDE + c4 * 4]) = v;
    }
    // ---- stage B chunk: KC x TNC, pair-interleaved, column guard ----
    // 64*64/256 = 16 elements per thread
#pragma unroll
    for (int i = 0; i < (KC * TNC) / 256; ++i) {
      int j = tid + i * 256;
      int kk = j >> 6;                             // 0..63
      int c  = j & 63;
      int gc = col0 + c;
      int gcc = (gc < M) ? gc : (M - 1);
      float v = B[(size_t)(kb + kk) * (size_t)M + (size_t)gcc];
      BsP[(kk >> 1) * B_PSTRIDE + 2 * c + (kk & 1)] = (gc < M) ? v : 0.0f;
    }
    __syncthreads();

    // prefetch next A chunk rows while we compute (global_prefetch_b8)
    if (kb + KC < K) {
      int grow = row0 + 16 * w + r;
      grow = (grow < N) ? grow : (N - 1);
      __builtin_prefetch(A + (size_t)grow * (size_t)K + (size_t)(kb + KC), 0, 0);
    }

    // ---- compute on staged chunk ----
#pragma unroll
    for (int kq = 0; kq < KC; kq += 4) {
      v2f a;
      a[0] = Afrag[kq];
      a[1] = Afrag[kq + 1];
      const float* B0 = Bfrag + (kq >> 1) * B_PSTRIDE;
#pragma unroll
      for (int t = 0; t < 4; ++t) {
        v2f b = *reinterpret_cast<const v2f*>(B0 + 32 * t);
        acc[t] = __builtin_amdgcn_wmma_f32_16x16x4_f32(
            /*neg_a=*/false, a, /*neg_b=*/false, b,
            /*c_mod=*/(short)0, acc[t], /*reuse_a=*/false, /*reuse_b=*/false);
      }
    }
    __syncthreads();
  }

  // ---- store C tile (row & col guards) ----
  const int rbase = row0 + 16 * w + 8 * h;
#pragma unroll
  for (int t = 0; t < 4; ++t) {
    int c = col0 + 16 * t + r;
    if (c < M) {
#pragma unroll
      for (int j = 0; j < 8; ++j) {
        int rowj = rbase + j;
        if (rowj < N)
          C[(size_t)rowj * (size_t)M + (size_t)c] = acc[t][j];
      }
    }
  }
}

// ---------------- edge scatter-add (message passing) ----------------
// out[dst[e]][f] += P[src[e]][f] * norm_src[src[e]]   (float4 vectorized)
__global__ void gcn_scatter_add(const float* __restrict__ P, const int* __restrict__ src,
                                const int* __restrict__ dst, const float* __restrict__ ns,
                                float* __restrict__ out, int E, int F4) {
  int idx = blockIdx.x * blockDim.x + threadIdx.x;
  int total = E * F4;
  if (idx >= total) return;
  int e = idx / F4;
  int q = idx - e * F4;
  int s = src[e];
  int d = dst[e];
  float w = ns[s];
  const float4 v = reinterpret_cast<const float4*>(P + (size_t)s * (size_t)(F4 * 4))[q];
  float* o = out + (size_t)d * (size_t)(F4 * 4) + (size_t)(q * 4);
  atomicAdd(o + 0, v.x * w);
  atomicAdd(o + 1, v.y * w);
  atomicAdd(o + 2, v.z * w);
  atomicAdd(o + 3, v.w * w);
}

// ---------------- BN stats: per-column sum / sumsq of y = G[n][f]*nd[n] ----
__global__ void gcn_bn_stats(const float* __restrict__ G, const float* __restrict__ nd,
                             float* __restrict__ stats, int N, int F, int rowsPerBlock) {
  int f = threadIdx.x;
  int r0 = blockIdx.x * rowsPerBlock;
  int r1 = min(r0 + rowsPerBlock, N);
  float s = 0.0f, s2 = 0.0f;
  for (int r = r0; r < r1; ++r) {
    float y = G[(size_t)r * (size_t)F + (size_t)f] * nd[r];
    s += y;
    s2 += y * y;
  }
  atomicAdd(&stats[f], s);
  atomicAdd(&stats[F + f], s2);
}

// ---------------- fused norm_dst * BN-affine + ReLU (in place) ----------------
__global__ void gcn_bn_apply(float* __restrict__ G, const float* __restrict__ nd,
                             const float* __restrict__ stats,
                             const float* __restrict__ gamma, const float* __restrict__ beta,
                             int N, int F) {
  int idx = blockIdx.x * blockDim.x + threadIdx.x;
  int total = N * F;
  if (idx >= total) return;
  int n = idx / F;
  int f = idx - n * F;
  float invN = 1.0f / (float)N;
  float mean = stats[f] * invN;
  float var = stats[F + f] * invN - mean * mean;
  float y = G[idx] * nd[n];
  y = (y - mean) * rsqrtf(var + GCN_BN_EPS) * gamma[f] + beta[f];
  G[idx] = fmaxf(y, 0.0f);
}

// ---------------- final: out = out * norm_dst + bias ----------------
__global__ void gcn_final_bias(float* __restrict__ out, const float* __restrict__ nd,
                               const float* __restrict__ b, int N, int F) {
  int idx = blockIdx.x * blockDim.x + threadIdx.x;
  int total = N * F;
  if (idx >= total) return;
  int n = idx / F;
  int f = idx - n * F;
  out[idx] = out[idx] * nd[n] + b[f];
}

// ---------------------------------------------------------------------------
extern "C" void kernel_launch(void* const* d_in, const int* in_sizes, int n_in,
                              void* d_out, int out_size, void* d_ws, size_t ws_size,
                              hipStream_t stream) {
  const float* feat  = (const float*)d_in[0];
  const int*   src   = (const int*)d_in[1];
  const int*   dst   = (const int*)d_in[2];
  const float* W0    = (const float*)d_in[3];
  const float* W1    = (const float*)d_in[4];
  const float* W2    = (const float*)d_in[5];
  const float* b2    = (const float*)d_in[6];
  const float* g0    = (const float*)d_in[7];
  const float* beta0 = (const float*)d_in[8];
  const float* g1    = (const float*)d_in[9];
  const float* beta1 = (const float*)d_in[10];
  float* out = (float*)d_out;

  const int N = in_sizes[0] / GCN_IN_FEATS;   // 170000
  const int E = in_sizes[1];                  // 1200000
  const int K0 = GCN_IN_FEATS;                // 128
  const int H  = GCN_HIDDEN;                  // 256
  const int C  = GCN_CLASSES;                 // 40

  // workspace layout
  char* ws = (char*)d_ws;
  float* P  = (float*)ws;                                         // [N, H] projection
  float* G  = (float*)(ws + (size_t)N * H * sizeof(float));       // [N, H] aggregate
  float* ns = (float*)(ws + 2 * (size_t)N * H * sizeof(float));   // [N] norm_src
  float* nd = ns + N;                                             // [N] norm_dst
  float* stats = nd + N;                                          // [2*H]

  const int TB = 256;
  auto blocks = [](long long n, int tb) { return (unsigned)((n + tb - 1) / tb); };

  // --- degree norms ---
  gcn_zero_f32<<<blocks(2 * N, TB), TB, 0, stream>>>(ns, 2 * N);  // ns & nd contiguous
  gcn_degree_count<<<blocks(E, TB), TB, 0, stream>>>(src, dst, ns, nd, E);
  gcn_finalize_norm<<<blocks(2 * N, TB), TB, 0, stream>>>(ns, 2 * N);

  // ================= layer 0: feat @ W0 -> aggregate -> BN+ReLU =================
  {
    dim3 g((N + TM - 1) / TM, (H + TNC - 1) / TNC);
    gcn_gemm_wmma_f32<<<g, 256, 0, stream>>>(feat, W0, P, N, K0, H);
  }
  gcn_zero_f32<<<blocks((long long)N * H, TB), TB, 0, stream>>>(G, N * H);
  gcn_scatter_add<<<blocks((long long)E * (H / 4), TB), TB, 0, stream>>>(P, src, dst, ns, G, E, H / 4);
  gcn_zero_f32<<<blocks(2 * H, TB), TB, 0, stream>>>(stats, 2 * H);
  gcn_bn_stats<<<blocks(N, 128), H, 0, stream>>>(G, nd, stats, N, H, 128);
  gcn_bn_apply<<<blocks((long long)N * H, TB), TB, 0, stream>>>(G, nd, stats, g0, beta0, N, H);

  // ================= layer 1: G @ W1 -> aggregate -> BN+ReLU =================
  {
    dim3 g((N + TM - 1) / TM, (H + TNC - 1) / TNC);
    gcn_gemm_wmma_f32<<<g, 256, 0, stream>>>(G, W1, P, N, H, H);
  }
  gcn_zero_f32<<<blocks((long long)N * H, TB), TB, 0, stream>>>(G, N * H);
  gcn_scatter_add<<<blocks((long long)E * (H / 4), TB), TB, 0, stream>>>(P, src, dst, ns, G, E, H / 4);
  gcn_zero_f32<<<blocks(2 * H, TB), TB, 0, stream>>>(stats, 2 * H);
  gcn_bn_stats<<<blocks(N, 128), H, 0, stream>>>(G, nd, stats, N, H, 128);
  gcn_bn_apply<<<blocks((long long)N * H, TB), TB, 0, stream>>>(G, nd, stats, g1, beta1, N, H);

  // ================= layer 2: G @ W2 -> aggregate into d_out -> bias ===========
  {
    dim3 g((N + TM - 1) / TM, (C + TNC - 1) / TNC);
    gcn_gemm_wmma_f32<<<g, 256, 0, stream>>>(G, W2, P, N, H, C);
  }
  gcn_zero_f32<<<blocks((long long)N * C, TB), TB, 0, stream>>>(out, N * C);
  gcn_scatter_add<<<blocks((long long)E * (C / 4), TB), TB, 0, stream>>>(P, src, dst, ns, out, E, C / 4);
  gcn_final_bias<<<blocks((long long)N * C, TB), TB, 0, stream>>>(out, nd, b2, N, C);
}